// RandomResizedCrop_70076686401815
// MI455X (gfx1250) — compile-verified
//
#include <hip/hip_runtime.h>
#include <stdint.h>

#define BLOCK 256
#define OUT_PER_BLOCK 4096
#define LDS_FLOATS 4224   // >= OUT_PER_BLOCK*step(<=1) + alignment + safety margins

typedef int   v4i __attribute__((ext_vector_type(4)));
typedef float v4f __attribute__((ext_vector_type(4)));
typedef __attribute__((address_space(1))) v4i as1_v4i;
typedef __attribute__((address_space(3))) v4i as3_v4i;

__global__ __launch_bounds__(BLOCK)
void RandomResizedCrop_resample_kernel(const float* __restrict__ audio,
                                       const int*   __restrict__ crop_len_p,
                                       const int*   __restrict__ start_pos_p,
                                       float*       __restrict__ out,
                                       int original_len)
{
    __shared__ float tile[LDS_FLOATS];

    const int crop_len  = crop_len_p[0];
    const int start_pos = start_pos_p[0];
    const int crop_m1   = crop_len - 1;

    const double step_d = (double)crop_m1 / (double)(original_len - 1);
    const float  step_f = (float)step_d;

    const long long out0  = (long long)blockIdx.x * OUT_PER_BLOCK;
    long long outLast     = out0 + OUT_PER_BLOCK - 1;
    if (outLast > original_len - 1) outLast = original_len - 1;

    // Conservative staged input window (margins absorb f32 index rounding, up to ~3)
    long long g0 = (long long)__builtin_floor((double)out0 * step_d) - 8;
    if (g0 < 0) g0 = 0;
    long long g1 = (long long)__builtin_floor((double)outLast * step_d) + 8;
    if (g1 > crop_m1) g1 = crop_m1;

    // Align the *absolute* source address to 16B for B128 async loads.
    const int absStart = (start_pos + (int)g0) & ~3;   // absolute float index into audio
    const int base     = absStart - start_pos;          // cropped-space index of tile[0]
    int span    = (int)(g1 - base) + 1;
    int nChunks = (span + 3) >> 2;                      // 16B chunks
    if (nChunks > LDS_FLOATS / 4) nChunks = LDS_FLOATS / 4;

    const float* gsrc = audio + absStart;               // 16B aligned

    // ---- Stage input window into LDS via CDNA5 async global->LDS copies ----
#if __has_builtin(__builtin_amdgcn_global_load_async_to_lds_b128)
    for (int c = threadIdx.x; c < nChunks; c += BLOCK) {
        as1_v4i* gp = (as1_v4i*)(uintptr_t)(gsrc + 4 * c);
        as3_v4i* lp = (as3_v4i*)(uint32_t)(uintptr_t)(tile + 4 * c);
        __builtin_amdgcn_global_load_async_to_lds_b128(gp, lp, 0, 0);
    }
#else
    for (int c = threadIdx.x; c < nChunks; c += BLOCK) {
        unsigned long long gaddr = (unsigned long long)(uintptr_t)(gsrc + 4 * c);
        unsigned           laddr = (unsigned)(uintptr_t)(tile + 4 * c); // low 32 bits = LDS offset
        asm volatile("global_load_async_to_lds_b128 %0, %1, off"
                     :: "v"(laddr), "v"(gaddr) : "memory");
    }
#endif
#if __has_builtin(__builtin_amdgcn_s_wait_asynccnt)
    __builtin_amdgcn_s_wait_asynccnt(0);
#else
    asm volatile("s_wait_asynccnt 0x0" ::: "memory");
#endif
    __syncthreads();

    // ---- Interpolate from LDS, store coalesced non-temporal float4 ----
    const int tid = threadIdx.x;
    const long long out4base = (long long)blockIdx.x * (OUT_PER_BLOCK / 4);

#pragma unroll
    for (int k = 0; k < 4; ++k) {
        const long long o4 = out4base + (long long)k * BLOCK + tid;
        const int i0 = (int)(o4 << 2);
        if (i0 >= original_len) continue;

        float r[4];
#pragma unroll
        for (int e = 0; e < 4; ++e) {
            const int   i   = i0 + e;
            const float idx = (float)i * step_f;       // matches f32 linspace semantics
            const float fl  = floorf(idx);
            const int   lo  = (int)fl;
            const float w   = idx - fl;                // reference: w from unclamped lo
            const int   loc = lo < crop_m1 ? lo : crop_m1;       // JAX clips OOB gather
            const int   hic = (lo + 1) < crop_m1 ? (lo + 1) : crop_m1;
            const float a   = tile[loc - base];
            const float b   = tile[hic - base];
            r[e] = (1.0f - w) * a + w * b;
        }

        if (i0 + 3 < original_len) {
            // Write-once stream: non-temporal B128 store (TH=NT) keeps L2 for the read stream.
            v4f v = { r[0], r[1], r[2], r[3] };
            __builtin_nontemporal_store(v, (v4f*)out + o4);
        } else {
            for (int e = 0; e < 4 && (i0 + e) < original_len; ++e) out[i0 + e] = r[e];
        }
    }
}

extern "C" void kernel_launch(void* const* d_in, const int* in_sizes, int n_in,
                              void* d_out, int out_size, void* d_ws, size_t ws_size,
                              hipStream_t stream) {
    (void)n_in; (void)out_size; (void)d_ws; (void)ws_size;
    const float* audio = (const float*)d_in[0];
    const int*   crop  = (const int*)d_in[1];
    const int*   start = (const int*)d_in[2];
    float*       out   = (float*)d_out;

    const int original_len = in_sizes[0];
    const int grid = (original_len + OUT_PER_BLOCK - 1) / OUT_PER_BLOCK;

    hipLaunchKernelGGL(RandomResizedCrop_resample_kernel,
                       dim3(grid), dim3(BLOCK), 0, stream,
                       audio, crop, start, out, original_len);
}